// CuboidCenterHead_62938450755677
// MI455X (gfx1250) — compile-verified
//
#include <hip/hip_runtime.h>

#ifndef __has_builtin
#define __has_builtin(x) 0
#endif

// CDNA5 async global->LDS path (ASYNCcnt-tracked), guarded so the file
// compiles even if this toolchain doesn't expose the builtin.
#if defined(__HIP_DEVICE_COMPILE__) && __has_builtin(__builtin_amdgcn_global_load_async_to_lds_b128)
#define USE_ASYNC_LDS 1
#else
#define USE_ASYNC_LDS 0
#endif

#if USE_ASYNC_LDS
// Builtin signature (from hipcc diagnostic): param 1 is
//   __attribute__((vector_size(16))) int  addrspace(1)*   (global src)
// so feed it exactly that; LDS dest cast to the AS(3) flavor.
typedef int v4i_t __attribute__((vector_size(16)));
#define AS1_V4I(p) ((__attribute__((address_space(1))) v4i_t*)(void*)(p))
#define AS3_V4I(p) ((__attribute__((address_space(3))) v4i_t*)(void*)(p))
#endif

#define BATCH 32
#define GX 128
#define GY 128
#define GZ 64
#define TX 8
#define TY 8
#define TOPK 10
#define BLOCKS_PER_BATCH 256 /* (GX/TX)*(GY/TY) = 16*16 */

__device__ __forceinline__ unsigned long long umax64(unsigned long long a,
                                                     unsigned long long b) {
  return a > b ? a : b;
}

// ---------------------------------------------------------------------------
// Phase 1: per 8x8x64 tile -> 3x3x3 max-pool NMS -> block-local top-10 keys.
// key = (float_bits(value) << 32) | ~flat_idx   (0 == "no candidate")
// ---------------------------------------------------------------------------
__global__ void __launch_bounds__(256)
nms_phase1(const float* __restrict__ src0, unsigned long long* __restrict__ ws) {
  __shared__ float s_in[10 * 10 * 64];  // halo tile (25.6 KB)
  __shared__ float s_zm[10 * 10 * 64];  // z-direction 3-max  (25.6 KB)
  __shared__ unsigned long long s_red[8];
  __shared__ unsigned long long s_win;

  const int t = threadIdx.x;
  const int ix0 = blockIdx.x * TX;
  const int iy0 = blockIdx.y * TY;
  const int b = blockIdx.z;
  const float* src = src0 + (size_t)b * (GX * GY * GZ);

  // ---- load 10x10x64 halo tile (1600 x 16B chunks), -inf outside volume ----
  const float NEG = -__builtin_inff();
  for (int c = t; c < 1600; c += 256) {
    int tix = c / 160;              // 0..9  (x within halo tile)
    int rem = c - tix * 160;        // 0..159
    int tiy = rem >> 4;             // 0..9  (y within halo tile)
    int z4 = (rem & 15) << 2;       // 0,4,...,60
    int gix = ix0 - 1 + tix;
    int giy = iy0 - 1 + tiy;
    float4* dst = (float4*)&s_in[(tix * 10 + tiy) * 64 + z4];
    if ((unsigned)gix < (unsigned)GX && (unsigned)giy < (unsigned)GY) {
      const float4* g =
          (const float4*)(src + ((size_t)(gix * GY + giy) * GZ + z4));
#if USE_ASYNC_LDS
      __builtin_amdgcn_global_load_async_to_lds_b128(AS1_V4I(g), AS3_V4I(dst),
                                                     0, 0);
#else
      *dst = *g;
#endif
    } else {
      *dst = make_float4(NEG, NEG, NEG, NEG);
    }
  }
#if USE_ASYNC_LDS
#if __has_builtin(__builtin_amdgcn_s_wait_asynccnt)
  __builtin_amdgcn_s_wait_asynccnt(0);
#else
  asm volatile("s_wait_asynccnt 0" ::: "memory");
#endif
#endif
  __syncthreads();

  // ---- separable step: 3-max along z for all 100 halo columns ----
  for (int e = t; e < 6400; e += 256) {  // 25 iterations
    int z = e & 63;
    int row = e >> 6;  // tix*10 + tiy, 0..99
    float m = s_in[row * 64 + z];
    if (z > 0) m = fmaxf(m, s_in[row * 64 + z - 1]);
    if (z < 63) m = fmaxf(m, s_in[row * 64 + z + 1]);
    s_zm[row * 64 + z] = m;
  }
  __syncthreads();

  // ---- peak test: x == max over 3x3 neighborhood of z-max columns ----
  unsigned long long keys[16];
#pragma unroll
  for (int i = 0; i < 16; i++) {
    int v = i * 256 + t;  // 0..4095 over the 8x8x64 tile
    int z = v & 63;
    int oy = (v >> 6) & 7;
    int ox = v >> 9;
    float xv = s_in[((ox + 1) * 10 + (oy + 1)) * 64 + z];
    float m = s_zm[(ox * 10 + oy) * 64 + z];
    m = fmaxf(m, s_zm[(ox * 10 + oy + 1) * 64 + z]);
    m = fmaxf(m, s_zm[(ox * 10 + oy + 2) * 64 + z]);
    m = fmaxf(m, s_zm[((ox + 1) * 10 + oy) * 64 + z]);
    m = fmaxf(m, s_zm[((ox + 1) * 10 + oy + 1) * 64 + z]);
    m = fmaxf(m, s_zm[((ox + 1) * 10 + oy + 2) * 64 + z]);
    m = fmaxf(m, s_zm[((ox + 2) * 10 + oy) * 64 + z]);
    m = fmaxf(m, s_zm[((ox + 2) * 10 + oy + 1) * 64 + z]);
    m = fmaxf(m, s_zm[((ox + 2) * 10 + oy + 2) * 64 + z]);
    unsigned idx = (unsigned)(((ix0 + ox) * GY + (iy0 + oy)) * GZ + z);
    unsigned long long key =
        ((unsigned long long)__float_as_uint(xv) << 32) | (unsigned)(~idx);
    keys[i] = (xv == m) ? key : 0ULL;
  }

  // ---- 10 rounds of block-wide argmax extraction ----
  unsigned long long* blk_out =
      ws + ((size_t)b * BLOCKS_PER_BATCH + blockIdx.y * 16 + blockIdx.x) * TOPK;
  for (int k = 0; k < TOPK; k++) {
    unsigned long long lm = 0ULL;
#pragma unroll
    for (int i = 0; i < 16; i++) lm = umax64(lm, keys[i]);
#pragma unroll
    for (int off = 16; off >= 1; off >>= 1)
      lm = umax64(lm, (unsigned long long)__shfl_xor(lm, off, 32));
    if ((t & 31) == 0) s_red[t >> 5] = lm;
    __syncthreads();
    if (t == 0) {
      unsigned long long w = s_red[0];
#pragma unroll
      for (int j = 1; j < 8; j++) w = umax64(w, s_red[j]);
      s_win = w;
      blk_out[k] = w;
    }
    __syncthreads();
    unsigned long long w = s_win;
    if (w) {
#pragma unroll
      for (int i = 0; i < 16; i++)
        if (keys[i] == w) keys[i] = 0ULL;
    }
  }
}

// ---------------------------------------------------------------------------
// Phase 2: one block per batch, merge 256 blocks x 10 keys -> final top-10,
// decode flat idx -> (ix,iy,iz) -> metric location, write [10][5] rows.
// ---------------------------------------------------------------------------
__global__ void __launch_bounds__(256)
nms_phase2(const unsigned long long* __restrict__ ws, float* __restrict__ out) {
  __shared__ unsigned long long s_red[8];
  __shared__ unsigned long long s_win;
  const int t = threadIdx.x;
  const int b = blockIdx.x;
  const unsigned long long* src = ws + (size_t)b * (BLOCKS_PER_BATCH * TOPK);

  unsigned long long keys[TOPK];
#pragma unroll
  for (int i = 0; i < TOPK; i++) keys[i] = src[i * 256 + t];

  for (int k = 0; k < TOPK; k++) {
    unsigned long long lm = 0ULL;
#pragma unroll
    for (int i = 0; i < TOPK; i++) lm = umax64(lm, keys[i]);
#pragma unroll
    for (int off = 16; off >= 1; off >>= 1)
      lm = umax64(lm, (unsigned long long)__shfl_xor(lm, off, 32));
    if ((t & 31) == 0) s_red[t >> 5] = lm;
    __syncthreads();
    if (t == 0) {
      unsigned long long w = s_red[0];
#pragma unroll
      for (int j = 1; j < 8; j++) w = umax64(w, s_red[j]);
      s_win = w;
      // decode + write output row [b][k][0..4]
      float val = __uint_as_float((unsigned)(w >> 32));
      unsigned idx = (~(unsigned)(w & 0xFFFFFFFFu)) & 0xFFFFFu;  // X*Y*Z = 2^20
      int ixg = (int)(idx >> 13);        // / (128*64)
      int iyg = (int)((idx >> 6) & 127); // / 64 % 128
      int izg = (int)(idx & 63);
      float* o = out + ((size_t)b * TOPK + k) * 5;
      o[0] = (float)ixg / 127.0f * 8000.0f - 4000.0f;
      o[1] = (float)iyg / 127.0f * 8000.0f - 4000.0f;
      o[2] = (float)izg / 63.0f * 2000.0f;  // +1000 - 1000
      o[3] = 0.0f;
      o[4] = val;
    }
    __syncthreads();
    unsigned long long w = s_win;
    if (w) {
#pragma unroll
      for (int i = 0; i < TOPK; i++)
        if (keys[i] == w) keys[i] = 0ULL;
    }
  }
}

extern "C" void kernel_launch(void* const* d_in, const int* in_sizes, int n_in,
                              void* d_out, int out_size, void* d_ws,
                              size_t ws_size, hipStream_t stream) {
  (void)in_sizes;
  (void)n_in;
  (void)out_size;
  (void)ws_size;  // needs 32*256*10*8 = 655,360 bytes of scratch
  const float* root = (const float*)d_in[0];
  float* out = (float*)d_out;
  unsigned long long* ws = (unsigned long long*)d_ws;

  dim3 g1(GX / TX, GY / TY, BATCH);  // 16 x 16 x 32
  nms_phase1<<<g1, 256, 0, stream>>>(root, ws);
  nms_phase2<<<BATCH, 256, 0, stream>>>(ws, out);
}